// DecoderLSTM_3607772529401
// MI455X (gfx1250) — compile-verified
//
#include <hip/hip_runtime.h>
#include <math.h>

// Problem dimensions (fixed by the reference)
#define Hh 512
#define Ee 512
#define Vv 10000
#define Ff 1280
#define Pp 196
#define Bb 32
#define Tt 80
#define G4 2048            // 4*H
#define XK 1792            // E + F
#define KCH 64             // K-chunk staged in LDS
#define LDSTR 68           // padded LDS row stride (floats), kills bank conflicts

typedef float v2f __attribute__((ext_vector_type(2)));
typedef float v8f __attribute__((ext_vector_type(8)));

__device__ __forceinline__ float sigmoidf_(float x) { return 1.0f / (1.0f + expf(-x)); }

// ---------------------------------------------------------------------------
// WMMA GEMM: D[M,N] = X[M,K] * W[N,K]^T (+ bias[n]) (+= if accFlag)
// Each wave computes a 16x32 output tile (two independent WMMA accumulator
// chains -> XDL pipeline gets independent work, A fragments reused 2x).
// grid.y = m-tile (16 rows), grid.x = group of 8 n-tiles, block = 128 (4 waves)
// K must be a multiple of 64; M a multiple of 16. nTiles may be ragged.
// ---------------------------------------------------------------------------
__global__ __launch_bounds__(128)
void gemm_xwt(const float* __restrict__ X, int ldX,
              const float* __restrict__ W, int ldW,
              const float* __restrict__ bias,
              float* __restrict__ D, long long ldD,
              int K, int nTiles, int accFlag)
{
    __shared__ float lA[16 * LDSTR];
    __shared__ float lB[4][32 * LDSTR];

    const int tid   = threadIdx.x;
    const int wave  = tid >> 5;
    const int lane  = tid & 31;
    const int m0    = blockIdx.y * 16;
    const int tile0 = (blockIdx.x * 4 + wave) * 2;  // wave's first n-tile
    const bool v0   = tile0 < nTiles;
    const bool v1   = tile0 + 1 < nTiles;
    const int nrow0 = tile0 * 16;                   // wave's first W row
    const int Nrows = nTiles * 16;

    const int row = lane & 15;            // shared row mapping for A/B fragments
    const int khi = (lane >> 4) << 1;     // 0 (lanes 0-15) or 2 (lanes 16-31)

    v8f acc0 = {};
    v8f acc1 = {};

    for (int k0 = 0; k0 < K; k0 += KCH) {
        // Cooperative A tile load: 16x64 floats as 256 float4, 128 threads
        #pragma unroll
        for (int r = 0; r < 2; ++r) {
            int e   = tid + r * 128;
            int ar  = e >> 4;
            int ac4 = e & 15;
            *(float4*)(&lA[ar * LDSTR + ac4 * 4]) =
                *(const float4*)(X + (size_t)(m0 + ar) * ldX + k0 + ac4 * 4);
        }
        // Per-wave B slab: 32 consecutive W rows (2 n-tiles), coalesced
        if (v0) {
            #pragma unroll
            for (int r = 0; r < 16; ++r) {
                int e   = lane + r * 32;            // 0..511 float4
                int br  = e >> 4;                   // row 0..31
                int bc4 = e & 15;
                if (nrow0 + br < Nrows) {
                    *(float4*)(&lB[wave][br * LDSTR + bc4 * 4]) =
                        *(const float4*)(W + (size_t)(nrow0 + br) * ldW + k0 + bc4 * 4);
                }
            }
            if (k0 + KCH < K)   // hint next weight chunk into cache (global_prefetch)
                __builtin_prefetch(W + (size_t)(nrow0 + row) * ldW + k0 + KCH, 0, 1);
        }
        __syncthreads();

        if (v0) {
            #pragma unroll
            for (int kk = 0; kk < KCH; kk += 4) {
                v2f a, b0, b1;
                a.x  = lA[row * LDSTR + kk + khi];
                a.y  = lA[row * LDSTR + kk + khi + 1];
                b0.x = lB[wave][row * LDSTR + kk + khi];
                b0.y = lB[wave][row * LDSTR + kk + khi + 1];
                b1.x = lB[wave][(row + 16) * LDSTR + kk + khi];
                b1.y = lB[wave][(row + 16) * LDSTR + kk + khi + 1];
                acc0 = __builtin_amdgcn_wmma_f32_16x16x4_f32(
                         false, a, false, b0, (short)0, acc0, false, false);
                acc1 = __builtin_amdgcn_wmma_f32_16x16x4_f32(
                         false, a, false, b1, (short)0, acc1, false, false);
            }
        }
        __syncthreads();
    }

    const int mbase = m0 + ((lane >> 4) << 3);   // +0 or +8
    if (v0) {
        const int nc   = nrow0 + row;
        const float bv = bias ? bias[nc] : 0.0f;
        #pragma unroll
        for (int j = 0; j < 8; ++j) {
            float r = acc0[j] + bv;
            float* dp = D + (size_t)(mbase + j) * (size_t)ldD + nc;
            if (accFlag) r += *dp;
            *dp = r;
        }
    }
    if (v1) {
        const int nc   = nrow0 + 16 + row;
        const float bv = bias ? bias[nc] : 0.0f;
        #pragma unroll
        for (int j = 0; j < 8; ++j) {
            float r = acc1[j] + bv;
            float* dp = D + (size_t)(mbase + j) * (size_t)ldD + nc;
            if (accFlag) r += *dp;
            *dp = r;
        }
    }
}

// ---------------------------------------------------------------------------
// One-time init kernels
// ---------------------------------------------------------------------------
__global__ void mean_enc_kernel(const float* __restrict__ feat, float* __restrict__ mean)
{
    int b = blockIdx.y;
    int f = blockIdx.x * 256 + threadIdx.x;          // F = 5*256
    const float* fp = feat + (size_t)b * Pp * Ff + f;
    float s = 0.f;
    for (int p = 0; p < Pp; ++p) s += fp[(size_t)p * Ff];
    mean[b * Ff + f] = s * (1.0f / (float)Pp);
}

__global__ void bias_sum_kernel(const float* __restrict__ a, const float* __restrict__ b,
                                float* __restrict__ o)
{
    int i = blockIdx.x * 256 + threadIdx.x;          // 4H = 8*256
    o[i] = a[i] + b[i];
}

__global__ void sos_kernel(const float* __restrict__ emb, float* __restrict__ xbuf)
{
    int b = blockIdx.y;
    int e = blockIdx.x * 256 + threadIdx.x;          // E = 2*256
    xbuf[(size_t)b * XK + e] = emb[(size_t)1 * Ee + e];   // SOS == 1
}

// ---------------------------------------------------------------------------
// Attention: scores[b,p] = sum_h relu(hid[b,h] + enc_proj[b,p,h]) * Va[h] + Va_b
// ---------------------------------------------------------------------------
__global__ __launch_bounds__(128)
void attn_scores_kernel(const float* __restrict__ hid, const float* __restrict__ encp,
                        const float* __restrict__ Va_w, const float* __restrict__ Va_b,
                        float* __restrict__ scores)
{
    int p = blockIdx.x, b = blockIdx.y, tid = threadIdx.x;
    const float* hp = hid + (size_t)b * Hh;
    const float* ep = encp + ((size_t)b * Pp + p) * Hh;
    float s = 0.f;
    for (int h = tid; h < Hh; h += 128)
        s += fmaxf(hp[h] + ep[h], 0.0f) * Va_w[h];
    __shared__ float red[128];
    red[tid] = s; __syncthreads();
    for (int st = 64; st > 0; st >>= 1) {
        if (tid < st) red[tid] += red[tid + st];
        __syncthreads();
    }
    if (tid == 0) scores[b * Pp + p] = red[0] + Va_b[0];
}

// softmax over P per batch row; writes weights (ws) and attentions (d_out slice)
__global__ __launch_bounds__(256)
void softmax_kernel(const float* __restrict__ scores, float* __restrict__ wts,
                    float* __restrict__ attnOut /* out+ATT+t*P */)
{
    int b = blockIdx.x, tid = threadIdx.x;
    __shared__ float red[256];
    float v = (tid < Pp) ? scores[b * Pp + tid] : -INFINITY;
    red[tid] = v; __syncthreads();
    for (int st = 128; st > 0; st >>= 1) {
        if (tid < st) red[tid] = fmaxf(red[tid], red[tid + st]);
        __syncthreads();
    }
    float mx = red[0]; __syncthreads();
    float e = (tid < Pp) ? expf(v - mx) : 0.0f;
    red[tid] = e; __syncthreads();
    for (int st = 128; st > 0; st >>= 1) {
        if (tid < st) red[tid] += red[tid + st];
        __syncthreads();
    }
    float inv = 1.0f / red[0];
    if (tid < Pp) {
        float w = e * inv;
        wts[b * Pp + tid] = w;
        attnOut[(size_t)b * Tt * Pp + tid] = w;
    }
}

// context[b,f] = (sum_p feat[b,p,f] * w[b,p]) * sigmoid(gateraw[b,f]) -> x[b, E+f]
__global__ __launch_bounds__(256)
void context_kernel(const float* __restrict__ feat, const float* __restrict__ wts,
                    const float* __restrict__ gateraw, float* __restrict__ xbuf)
{
    int b = blockIdx.y, tid = threadIdx.x;
    int f = blockIdx.x * 256 + tid;                  // F = 5*256
    __shared__ float w[Pp];
    if (tid < Pp) w[tid] = wts[b * Pp + tid];
    __syncthreads();
    const float* fp = feat + (size_t)b * Pp * Ff + f;
    float acc = 0.f;
    for (int p = 0; p < Pp; ++p) acc += fp[(size_t)p * Ff] * w[p];
    float g = sigmoidf_(gateraw[b * Ff + f]);
    xbuf[(size_t)b * XK + Ee + f] = acc * g;
}

// LSTM cell pointwise: gate order i,f,g,o
__global__ __launch_bounds__(256)
void lstm_pointwise_kernel(const float* __restrict__ gates,
                           float* __restrict__ h, float* __restrict__ c)
{
    int b = blockIdx.y;
    int k = blockIdx.x * 256 + threadIdx.x;          // H = 2*256
    const float* g = gates + (size_t)b * G4;
    float ig = sigmoidf_(g[k]);
    float fg = sigmoidf_(g[Hh + k]);
    float gg = tanhf(g[2 * Hh + k]);
    float og = sigmoidf_(g[3 * Hh + k]);
    float cn = fg * c[b * Hh + k] + ig * gg;
    c[b * Hh + k] = cn;
    h[b * Hh + k] = og * tanhf(cn);
}

// greedy argmax over logits row, then gather embedding row into x[:, :E]
__global__ __launch_bounds__(256)
void argmax_embed_kernel(const float* __restrict__ logits /* out + t*V */,
                         const float* __restrict__ emb, float* __restrict__ xbuf)
{
    int b = blockIdx.x, tid = threadIdx.x;
    const float* lp = logits + (size_t)b * Tt * Vv;
    float best = -INFINITY; int bi = 0;
    for (int v = tid; v < Vv; v += 256) {
        float x = lp[v];
        if (x > best) { best = x; bi = v; }
    }
    __shared__ float sv[256];
    __shared__ int   si[256];
    sv[tid] = best; si[tid] = bi; __syncthreads();
    for (int st = 128; st > 0; st >>= 1) {
        if (tid < st) {
            float ov = sv[tid + st]; int oi = si[tid + st];
            if (ov > sv[tid] || (ov == sv[tid] && oi < si[tid])) { sv[tid] = ov; si[tid] = oi; }
        }
        __syncthreads();
    }
    int widx = si[0];
    for (int e = tid; e < Ee; e += 256)
        xbuf[(size_t)b * XK + e] = emb[(size_t)widx * Ee + e];
}

__global__ void copy_hc_kernel(const float* __restrict__ h, const float* __restrict__ c,
                               float* __restrict__ dst)
{
    int i = blockIdx.x * 256 + threadIdx.x;          // 2*B*H = 128*256
    if (i < Bb * Hh) dst[i] = h[i];
    else             dst[i] = c[i - Bb * Hh];
}

// ---------------------------------------------------------------------------
extern "C" void kernel_launch(void* const* d_in, const int* in_sizes, int n_in,
                              void* d_out, int out_size, void* d_ws, size_t ws_size,
                              hipStream_t stream)
{
    const float* feat   = (const float*)d_in[0];     // [B,P,F]
    const float* emb    = (const float*)d_in[3];     // [V,E]
    const float* W1_w   = (const float*)d_in[4];     // [H,H]
    const float* W1_b   = (const float*)d_in[5];
    const float* W2_w   = (const float*)d_in[6];     // [H,F]
    const float* W2_b   = (const float*)d_in[7];
    const float* Va_w   = (const float*)d_in[8];     // [1,H]
    const float* Va_b   = (const float*)d_in[9];
    const float* Wih    = (const float*)d_in[10];    // [4H, E+F]
    const float* Whh    = (const float*)d_in[11];    // [4H, H]
    const float* bih    = (const float*)d_in[12];
    const float* bhh    = (const float*)d_in[13];
    const float* out_w  = (const float*)d_in[14];    // [V,H]
    const float* out_b  = (const float*)d_in[15];
    const float* ih_w   = (const float*)d_in[16];    // [H,F]
    const float* ih_b   = (const float*)d_in[17];
    const float* ic_w   = (const float*)d_in[18];    // [H,F]
    const float* ic_b   = (const float*)d_in[19];
    const float* gate_w = (const float*)d_in[20];    // [F,H]
    const float* gate_b = (const float*)d_in[21];

    float* out = (float*)d_out;
    const size_t HB  = (size_t)Bb * Tt * Vv;         // h goes here
    const size_t CB  = HB + (size_t)Bb * Hh;         // then c
    const size_t ATT = CB + (size_t)Bb * Hh;         // then attentions [B,T,P]

    // Workspace layout (floats)
    float* ws      = (float*)d_ws;
    float* h       = ws;                             // B*H
    float* c       = h + Bb * Hh;                    // B*H
    float* hid     = c + Bb * Hh;                    // B*H
    float* xbuf    = hid + Bb * Hh;                  // B*XK
    float* gates   = xbuf + (size_t)Bb * XK;         // B*4H
    float* gateraw = gates + (size_t)Bb * G4;        // B*F
    float* scores  = gateraw + (size_t)Bb * Ff;      // B*P
    float* wts     = scores + Bb * Pp;               // B*P
    float* meanenc = wts + Bb * Pp;                  // B*F
    float* bsum    = meanenc + (size_t)Bb * Ff;      // 4H
    float* encp    = bsum + G4;                      // (B*P)*H

    // n-tile group counts (8 n-tiles per block)
    const int gH  = (Hh / 16 + 7) / 8;               // 4
    const int gF  = (Ff / 16 + 7) / 8;               // 10
    const int gG4 = (G4 / 16 + 7) / 8;               // 16
    const int gV  = (Vv / 16 + 7) / 8;               // 79

    // ---- one-time init ----
    mean_enc_kernel<<<dim3(Ff / 256, Bb), 256, 0, stream>>>(feat, meanenc);
    bias_sum_kernel<<<dim3(G4 / 256), 256, 0, stream>>>(bih, bhh, bsum);
    sos_kernel<<<dim3(Ee / 256, Bb), 256, 0, stream>>>(emb, xbuf);
    // h0 = mean @ init_h_w^T + b ; c0 likewise
    gemm_xwt<<<dim3(gH, 2), 128, 0, stream>>>(meanenc, Ff, ih_w, Ff, ih_b, h, Hh, Ff, Hh / 16, 0);
    gemm_xwt<<<dim3(gH, 2), 128, 0, stream>>>(meanenc, Ff, ic_w, Ff, ic_b, c, Hh, Ff, Hh / 16, 0);
    // enc_proj[b,p,h] = feat @ W2^T + W2_b   (M = B*P = 6272)
    gemm_xwt<<<dim3(gH, (Bb * Pp) / 16), 128, 0, stream>>>(feat, Ff, W2_w, Ff, W2_b, encp, Hh, Ff, Hh / 16, 0);

    // ---- recurrent steps ----
    for (int t = 0; t < Tt; ++t) {
        // hid = h @ W1^T + W1_b
        gemm_xwt<<<dim3(gH, 2), 128, 0, stream>>>(h, Hh, W1_w, Hh, W1_b, hid, Hh, Hh, Hh / 16, 0);
        // attention scores + softmax (weights also land in attentions output)
        attn_scores_kernel<<<dim3(Pp, Bb), 128, 0, stream>>>(hid, encp, Va_w, Va_b, scores);
        softmax_kernel<<<dim3(Bb), 256, 0, stream>>>(scores, wts, out + ATT + (size_t)t * Pp);
        // gate_raw = h @ gate_w^T + gate_b
        gemm_xwt<<<dim3(gF, 2), 128, 0, stream>>>(h, Hh, gate_w, Hh, gate_b, gateraw, Ff, Hh, Ff / 16, 0);
        // gated context -> x[:, E:]
        context_kernel<<<dim3(Ff / 256, Bb), 256, 0, stream>>>(feat, wts, gateraw, xbuf);
        // gates = x @ Wih^T + (bih+bhh)  then  += h @ Whh^T
        gemm_xwt<<<dim3(gG4, 2), 128, 0, stream>>>(xbuf, XK, Wih, XK, bsum, gates, G4, XK, G4 / 16, 0);
        gemm_xwt<<<dim3(gG4, 2), 128, 0, stream>>>(h, Hh, Whh, Hh, nullptr, gates, G4, Hh, G4 / 16, 1);
        // LSTM cell update (h, c)
        lstm_pointwise_kernel<<<dim3(Hh / 256, Bb), 256, 0, stream>>>(gates, h, c);
        // logits = h_new @ out_w^T + out_b  -> decoder_outputs[:, t, :]
        gemm_xwt<<<dim3(gV, 2), 128, 0, stream>>>(
            h, Hh, out_w, Hh, out_b, out + (size_t)t * Vv, (long long)Tt * Vv, Hh, Vv / 16, 0);
        // greedy decode: argmax + embedding gather -> x[:, :E]
        argmax_embed_kernel<<<dim3(Bb), 256, 0, stream>>>(out + (size_t)t * Vv, emb, xbuf);
    }

    // final (h, c)
    copy_hc_kernel<<<dim3((2 * Bb * Hh) / 256), 256, 0, stream>>>(h, c, out + HB);
}